// MixedGNN_11974368821437
// MI455X (gfx1250) — compile-verified
//
#include <hip/hip_runtime.h>

// ---------------------------------------------------------------------------
// MixedGNN for MI455X (gfx1250, wave32, WMMA).
// msg = relu(h0[src] @ W_msg + b) depends only on src => precompute per-node
// M[v], init aggr = M (self loops); the edge phase becomes gather + hardware
// f32 scatter-atomics that stay resident in the 192MB L2 (h0/M/aggr = 77MB).
// HBM traffic ~ edge_index (102MB) + x_local (12.8MB) => ~5us floor @23.3TB/s;
// real limiter is L2 atomic throughput, so exactly one global_atomic_add_f32
// per (edge,feature), coalesced 128B per wave.
// All node-level GEMMs use V_WMMA_F32_16X16X4_F32 (16-node tiles per wave).
// ---------------------------------------------------------------------------

typedef float v2f __attribute__((ext_vector_type(2)));
typedef float v8f __attribute__((ext_vector_type(8)));

__device__ __forceinline__ v8f wmma4(v2f a, v2f b, v8f c) {
  // D(16x16) = A(16x4) * B(4x16) + C   (f32, wave32)
  return __builtin_amdgcn_wmma_f32_16x16x4_f32(
      /*neg_a=*/false, a, /*neg_b=*/false, b, /*c_mod=*/(short)0, c,
      /*reuse_a=*/false, /*reuse_b=*/false);
}

// A-fragment: 16x4 tile, rows in lanes 0..15 and 16..31 (same rows), K split:
//   lanes 0-15  -> K = k0, k0+1 ; lanes 16-31 -> K = k0+2, k0+3
// B-fragment: 4x16 tile (from [K,32] row-major in LDS), cols n0..n0+15 in r,
//   K rows split across lane halves mirroring A.
// D-fragment: VGPR j holds row (j + 8*half), col (n0 + r).

// ---------------------------------------------------------------------------
// Kernel 0: h_global_graph = relu(x_global[128,8] @ W_global[8,32] + b_global)
// ---------------------------------------------------------------------------
__global__ __launch_bounds__(256) void hg_kernel(
    const float* __restrict__ x_global, const float* __restrict__ W_global,
    const float* __restrict__ b_global, float* __restrict__ hg) {
  int i = blockIdx.x * 256 + threadIdx.x;
  if (i >= 128 * 32) return;
  int row = i >> 5, col = i & 31;
  float acc = b_global[col];
#pragma unroll
  for (int k = 0; k < 8; ++k) acc += x_global[row * 8 + k] * W_global[k * 32 + col];
  hg[i] = fmaxf(acc, 0.f);
}

// ---------------------------------------------------------------------------
// Kernel 1: encoder + per-node message.  4 waves/block, 16 nodes/wave.
//   h_local = relu(x_local @ W_local + b_local)
//   h_cat   = [h_local | h_global[batch] | h_local*h_global]
//   h0      = relu(h_cat @ W_mix + b_mix)            -> h0g
//   M       = relu(h0 @ W_msg + b_msg)               -> Mg, aggr (self loop)
// ---------------------------------------------------------------------------
__global__ __launch_bounds__(128) void encode_kernel(
    const float* __restrict__ x_local, const long long* __restrict__ batch,
    const float* __restrict__ hg,
    const float* __restrict__ W_local, const float* __restrict__ b_local,
    const float* __restrict__ W_mix, const float* __restrict__ b_mix,
    const float* __restrict__ W_msg, const float* __restrict__ b_msg,
    float* __restrict__ h0g, float* __restrict__ Mg, float* __restrict__ aggrg,
    int n_tiles) {
  __shared__ float sWl[16 * 32];
  __shared__ float sWm[96 * 32];
  __shared__ float sWg[32 * 32];
  __shared__ float sbl[32], sbm[32], sbg[32];
  __shared__ float bufA[4][16 * 96];  // per-wave h_cat staging (reused for M)
  __shared__ float bufB[4][16 * 32];  // per-wave h0 staging

  const int tid = threadIdx.x;
  for (int i = tid; i < 16 * 32; i += 128) sWl[i] = W_local[i];
  for (int i = tid; i < 96 * 32; i += 128) sWm[i] = W_mix[i];
  for (int i = tid; i < 32 * 32; i += 128) sWg[i] = W_msg[i];
  if (tid < 32) { sbl[tid] = b_local[tid]; sbm[tid] = b_mix[tid]; sbg[tid] = b_msg[tid]; }
  __syncthreads();

  const int w = tid >> 5, lane = tid & 31, r = lane & 15, half = lane >> 4;
  int tile = blockIdx.x * 4 + w;
  if (tile >= n_tiles) tile = n_tiles - 1;  // N is a multiple of 16; safe clamp
  const int node0 = tile << 4;

  float* A = bufA[w];
  float* Bq = bufB[w];

  // ---- h_local via WMMA, A-fragments straight from global (L2-cached) ----
  v8f c0 = {}, c1 = {};
#pragma unroll
  for (int k0 = 0; k0 < 16; k0 += 4) {
    const int ka = k0 + 2 * half;
    v2f a; a.x = x_local[(node0 + r) * 16 + ka]; a.y = x_local[(node0 + r) * 16 + ka + 1];
    v2f b0; b0.x = sWl[ka * 32 + r];      b0.y = sWl[(ka + 1) * 32 + r];
    v2f b1; b1.x = sWl[ka * 32 + 16 + r]; b1.y = sWl[(ka + 1) * 32 + 16 + r];
    c0 = wmma4(a, b0, c0);
    c1 = wmma4(a, b1, c1);
  }
  const float bl0 = sbl[r], bl1 = sbl[16 + r];
#pragma unroll
  for (int j = 0; j < 8; ++j) {
    c0[j] = fmaxf(c0[j] + bl0, 0.f);
    c1[j] = fmaxf(c1[j] + bl1, 0.f);
  }

  // ---- gather h_global, build h_cat rows in LDS (D-fragment scatter) ----
#pragma unroll
  for (int j = 0; j < 8; ++j) {
    const int row = j + 8 * half;
    const int gb = (int)batch[node0 + row];
    const float g0 = hg[gb * 32 + r];
    const float g1 = hg[gb * 32 + 16 + r];
    A[row * 96 + r]      = c0[j];
    A[row * 96 + 16 + r] = c1[j];
    A[row * 96 + 32 + r] = g0;
    A[row * 96 + 48 + r] = g1;
    A[row * 96 + 64 + r] = c0[j] * g0;
    A[row * 96 + 80 + r] = c1[j] * g1;
  }
  __syncthreads();

  // ---- h0 = relu(h_cat[16,96] @ W_mix[96,32] + b_mix) ----
  v8f d0 = {}, d1 = {};
#pragma unroll
  for (int k0 = 0; k0 < 96; k0 += 4) {
    const int ka = k0 + 2 * half;
    v2f a; a.x = A[r * 96 + ka]; a.y = A[r * 96 + ka + 1];
    v2f b0; b0.x = sWm[ka * 32 + r];      b0.y = sWm[(ka + 1) * 32 + r];
    v2f b1; b1.x = sWm[ka * 32 + 16 + r]; b1.y = sWm[(ka + 1) * 32 + 16 + r];
    d0 = wmma4(a, b0, d0);
    d1 = wmma4(a, b1, d1);
  }
  const float bm0 = sbm[r], bm1 = sbm[16 + r];
#pragma unroll
  for (int j = 0; j < 8; ++j) {
    d0[j] = fmaxf(d0[j] + bm0, 0.f);
    d1[j] = fmaxf(d1[j] + bm1, 0.f);
    const int row = j + 8 * half;
    Bq[row * 32 + r]      = d0[j];
    Bq[row * 32 + 16 + r] = d1[j];
  }
  __syncthreads();

  {  // coalesced b128 store of h0 tile
    const float4* s4 = (const float4*)Bq;
    float4* dst4 = (float4*)(h0g + (size_t)node0 * 32);
#pragma unroll
    for (int i = 0; i < 4; ++i) dst4[i * 32 + lane] = s4[i * 32 + lane];
  }

  // ---- M = relu(h0[16,32] @ W_msg[32,32] + b_msg) ----
  v8f e0 = {}, e1 = {};
#pragma unroll
  for (int k0 = 0; k0 < 32; k0 += 4) {
    const int ka = k0 + 2 * half;
    v2f a; a.x = Bq[r * 32 + ka]; a.y = Bq[r * 32 + ka + 1];
    v2f b0; b0.x = sWg[ka * 32 + r];      b0.y = sWg[(ka + 1) * 32 + r];
    v2f b1; b1.x = sWg[ka * 32 + 16 + r]; b1.y = sWg[(ka + 1) * 32 + 16 + r];
    e0 = wmma4(a, b0, e0);
    e1 = wmma4(a, b1, e1);
  }
  const float bg0 = sbg[r], bg1 = sbg[16 + r];
#pragma unroll
  for (int j = 0; j < 8; ++j) {
    e0[j] = fmaxf(e0[j] + bg0, 0.f);
    e1[j] = fmaxf(e1[j] + bg1, 0.f);
    const int row = j + 8 * half;
    A[row * 32 + r]      = e0[j];  // reuse bufA as M staging
    A[row * 32 + 16 + r] = e1[j];
  }
  __syncthreads();
  {  // coalesced store: M, and aggr initialized to M (self-loop message)
    const float4* s4 = (const float4*)A;
    float4* dM = (float4*)(Mg + (size_t)node0 * 32);
    float4* dA = (float4*)(aggrg + (size_t)node0 * 32);
#pragma unroll
    for (int i = 0; i < 4; ++i) {
      float4 v = s4[i * 32 + lane];
      dM[i * 32 + lane] = v;
      dA[i * 32 + lane] = v;
    }
  }
}

// ---------------------------------------------------------------------------
// Kernel 2: edge scatter.
// Per wave: load 32 edge pairs coalesced (one pair per lane), then for each
// edge broadcast src/dst to SGPRs via readlane; all 32 lanes gather M[src]
// (one coalesced 128B line from L2) and issue a hardware global_atomic_add_f32
// into aggr[dst] (no-return form, STOREcnt). Inline asm guarantees the fp
// atomic never falls back to a CAS loop. Prefetch the next index chunk.
// ---------------------------------------------------------------------------
__global__ __launch_bounds__(256) void edge_kernel(
    const long long* __restrict__ edge_index,  // [2, E] int64
    const float* __restrict__ Mg, float* __restrict__ aggrg, long long E) {
  const int lane = threadIdx.x & 31;
  const long long warpId = ((long long)blockIdx.x * 256 + threadIdx.x) >> 5;
  const long long nWarps = ((long long)gridDim.x * 256) >> 5;
  const long long chunkStride = nWarps * 32;

  for (long long e0 = warpId * 32; e0 < E; e0 += chunkStride) {
    long long eidx = e0 + lane;
    if (eidx >= E) eidx = E - 1;                 // clamp (tail-safe)
    const int se = (int)edge_index[eidx];        // coalesced b64
    const int de = (int)edge_index[E + eidx];    // coalesced b64

    // prefetch next chunk of indices (streams from HBM)
    if (e0 + chunkStride < E) {
      __builtin_prefetch(&edge_index[e0 + chunkStride + lane], 0, 1);
      __builtin_prefetch(&edge_index[E + e0 + chunkStride + lane], 0, 1);
    }

    const long long rem = E - e0;
    const int cnt = rem < 32 ? (int)rem : 32;    // wave-uniform

#pragma unroll
    for (int i = 0; i < 32; ++i) {
      if (i < cnt) {
        const int src = __builtin_amdgcn_readlane(se, i);
        const int dst = __builtin_amdgcn_readlane(de, i);
        const float v = Mg[(size_t)src * 32 + lane];
        float* p = aggrg + (size_t)dst * 32 + lane;
        asm volatile("global_atomic_add_f32 %0, %1, off"
                     :
                     : "v"(p), "v"(v)
                     : "memory");
      }
    }
  }
}

// ---------------------------------------------------------------------------
// Kernel 3: h = relu(aggr + h0 @ W_self + b_self); out = h @ W_out + b_out
// 8 waves/block, 16 nodes/wave.
// ---------------------------------------------------------------------------
__global__ __launch_bounds__(256) void final_kernel(
    const float* __restrict__ h0g, const float* __restrict__ aggrg,
    const float* __restrict__ W_self, const float* __restrict__ b_self,
    const float* __restrict__ W_out, const float* __restrict__ b_out,
    float* __restrict__ out, int n_tiles) {
  __shared__ float sWs[32 * 32];
  __shared__ float sWo[32 * 2];
  __shared__ float sbs[32], sbo[2];
  __shared__ float bufA[8][16 * 32];  // h0 tile (later reused for h)
  __shared__ float bufB[8][16 * 32];  // aggr tile

  const int tid = threadIdx.x;
  for (int i = tid; i < 32 * 32; i += 256) sWs[i] = W_self[i];
  if (tid < 64) sWo[tid] = W_out[tid];
  if (tid < 32) sbs[tid] = b_self[tid];
  if (tid < 2) sbo[tid] = b_out[tid];
  __syncthreads();

  const int w = tid >> 5, lane = tid & 31, r = lane & 15, half = lane >> 4;
  int tile = blockIdx.x * 8 + w;
  const bool valid = tile < n_tiles;
  if (!valid) tile = n_tiles - 1;  // keep control flow uniform for barriers
  const int node0 = tile << 4;

  float* A = bufA[w];
  float* G = bufB[w];
  {  // coalesced tile loads
    const float4* h4 = (const float4*)(h0g + (size_t)node0 * 32);
    const float4* a4 = (const float4*)(aggrg + (size_t)node0 * 32);
    float4* A4 = (float4*)A;
    float4* G4 = (float4*)G;
#pragma unroll
    for (int i = 0; i < 4; ++i) {
      A4[i * 32 + lane] = h4[i * 32 + lane];
      G4[i * 32 + lane] = a4[i * 32 + lane];
    }
  }
  __syncthreads();

  // ---- self term: h0 @ W_self ----
  v8f s0 = {}, s1 = {};
#pragma unroll
  for (int k0 = 0; k0 < 32; k0 += 4) {
    const int ka = k0 + 2 * half;
    v2f a; a.x = A[r * 32 + ka]; a.y = A[r * 32 + ka + 1];
    v2f b0; b0.x = sWs[ka * 32 + r];      b0.y = sWs[(ka + 1) * 32 + r];
    v2f b1; b1.x = sWs[ka * 32 + 16 + r]; b1.y = sWs[(ka + 1) * 32 + 16 + r];
    s0 = wmma4(a, b0, s0);
    s1 = wmma4(a, b1, s1);
  }
  const float bs0 = sbs[r], bs1 = sbs[16 + r];
  __syncthreads();  // all A-fragment reads retired before overwrite below
#pragma unroll
  for (int j = 0; j < 8; ++j) {
    const int row = j + 8 * half;
    A[row * 32 + r]      = fmaxf(G[row * 32 + r] + s0[j] + bs0, 0.f);
    A[row * 32 + 16 + r] = fmaxf(G[row * 32 + 16 + r] + s1[j] + bs1, 0.f);
  }
  __syncthreads();

  // ---- head: 16 nodes x 2 classes = one output per lane, coalesced store --
  {
    const int node = lane >> 1, c = lane & 1;
    float acc = sbo[c];
#pragma unroll
    for (int k = 0; k < 32; ++k) acc += A[node * 32 + k] * sWo[k * 2 + c];
    if (valid) out[(size_t)(node0 + node) * 2 + c] = acc;
  }
}

// ---------------------------------------------------------------------------
extern "C" void kernel_launch(void* const* d_in, const int* in_sizes, int n_in,
                              void* d_out, int out_size, void* d_ws,
                              size_t ws_size, hipStream_t stream) {
  const float* x_local      = (const float*)d_in[0];
  const float* x_global     = (const float*)d_in[1];
  const long long* batch    = (const long long*)d_in[2];
  const long long* edge_idx = (const long long*)d_in[3];
  const float* W_local = (const float*)d_in[4];
  const float* b_local = (const float*)d_in[5];
  const float* W_global = (const float*)d_in[6];
  const float* b_global = (const float*)d_in[7];
  const float* W_mix = (const float*)d_in[8];
  const float* b_mix = (const float*)d_in[9];
  const float* W_msg = (const float*)d_in[10];
  const float* b_msg = (const float*)d_in[11];
  const float* W_self = (const float*)d_in[12];
  const float* b_self = (const float*)d_in[13];
  const float* W_out = (const float*)d_in[14];
  const float* b_out = (const float*)d_in[15];

  const int N = in_sizes[0] / 16;                  // 200000
  const long long E = (long long)in_sizes[3] / 2;  // 6400000
  const int n_tiles = (N + 15) / 16;               // 12500

  float* hg = (float*)d_ws;                // 128*32
  float* h0 = hg + 128 * 32;               // N*32
  float* Mg = h0 + (size_t)N * 32;         // N*32
  float* ag = Mg + (size_t)N * 32;         // N*32

  hg_kernel<<<16, 256, 0, stream>>>(x_global, W_global, b_global, hg);
  encode_kernel<<<(n_tiles + 3) / 4, 128, 0, stream>>>(
      x_local, batch, hg, W_local, b_local, W_mix, b_mix, W_msg, b_msg,
      h0, Mg, ag, n_tiles);
  edge_kernel<<<8192, 256, 0, stream>>>(edge_idx, Mg, ag, E);
  final_kernel<<<(n_tiles + 7) / 8, 256, 0, stream>>>(
      h0, ag, W_self, b_self, W_out, b_out, (float*)d_out, n_tiles);
}